// TimeDomainModel_20469814133344
// MI455X (gfx1250) — compile-verified
//
#include <hip/hip_runtime.h>
#include <hip/hip_bf16.h>

// Balloon/Windkessel BOLD forward-Euler scan, N_STEPS = 1000.
// Sequential nonlinear scalar recurrence: single wave32, lane 0 computes.
// Pure latency problem (~25 KFLOP, 8 KB traffic). No matrix structure ->
// WMMA inapplicable. This revision removes every IEEE-divide sequence from
// the loop: fout*q/v_safe folds to q (fout==v_safe since alpha==1), E uses
// v_rcp_f32 + v_exp_f32 directly, y uses v_rcp_f32. Critical path per step
// is f(FMA) -> rcp -> mul -> exp2 -> fma(q).

#define BOLD_N_STEPS 1000
#define BOLD_DT      0.01f
#define BOLD_V0      0.04f

__global__ __launch_bounds__(32, 1)
void bold_scan_kernel(const float* __restrict__ p_mtt,
                      const float* __restrict__ p_sigma,
                      const float* __restrict__ p_mu,
                      const float* __restrict__ p_lamb,
                      const float* __restrict__ p_c,
                      const float* __restrict__ p_beta,
                      const float* __restrict__ u,
                      float* __restrict__ out)
{
    const int lane = threadIdx.x;
    // Pull all 4 KB of u_drive toward the WGP before the serial loop:
    // 32 lanes x 128 B = 4096 B (speculative; OOB tail dropped).
    __builtin_prefetch(u + lane * 32, /*rw=*/0, /*locality=*/3);
    if (lane != 0) return;

    const float mtt   = *p_mtt;
    const float sigma = *p_sigma;
    const float mu    = *p_mu;
    const float lamb  = *p_lamb;
    const float cc    = *p_c;
    const float beta  = *p_beta;

    // Hoisted constants (computed once; precise).
    const float L2          = log2f(1.0f - lamb);   // log2(1-lamb) < 0
    const float inv_lamb    = 1.0f / lamb;
    const float k1          = 7.0f * lamb;
    const float k3          = 2.0f * lamb - 0.2f;
    const float scale       = cc * beta * BOLD_V0;
    const float dt_over_mtt = BOLD_DT / mtt;

    float s = 0.0f, f = 1.0f, v = 1.0f, q = 1.0f;

    const float4* __restrict__ u4   = (const float4*)u;
    float4*       __restrict__ out4 = (float4*)out;

    #pragma unroll 1
    for (int i = 0; i < BOLD_N_STEPS / 4; ++i) {
        const float4 uu = u4[i];              // 4 steps of drive per load
        const float us[4] = {uu.x, uu.y, uu.z, uu.w};
        float ys[4];
        #pragma unroll
        for (int j = 0; j < 4; ++j) {
            const float u_t    = us[j];
            const float f_safe = fmaxf(f, 1e-3f);
            const float v_safe = fmaxf(v, 1e-3f);
            // E = 1 - (1-lamb)^(1/f_safe) = 1 - exp2(log2(1-lamb)/f_safe)
            const float E = 1.0f - __builtin_amdgcn_exp2f(
                                       L2 * __builtin_amdgcn_rcpf(f_safe));

            const float s_n = s + BOLD_DT * (u_t - sigma * s - mu * (f - 1.0f));
            const float f_n = f + BOLD_DT * s;
            // fout = v_safe ** (1/alpha) with alpha == 1  ->  fout = v_safe
            const float v_n = v + (f - v_safe) * dt_over_mtt;
            // fout*q/v_safe == q exactly (same numerator/denominator)
            const float q_n = q + (f * E * inv_lamb - q) * dt_over_mtt;

            const float v_o = fmaxf(v_n, 1e-3f);
            ys[j] = scale * (k1 * (1.0f - q_n)
                           + 2.0f * (1.0f - q_n * __builtin_amdgcn_rcpf(v_o))
                           + k3 * (1.0f - v_n));

            s = s_n; f = f_n; v = v_n; q = q_n;
        }
        out4[i] = make_float4(ys[0], ys[1], ys[2], ys[3]);
    }
}

extern "C" void kernel_launch(void* const* d_in, const int* in_sizes, int n_in,
                              void* d_out, int out_size, void* d_ws, size_t ws_size,
                              hipStream_t stream) {
    (void)in_sizes; (void)n_in; (void)d_ws; (void)ws_size; (void)out_size;
    const float* mtt   = (const float*)d_in[0];
    const float* sigma = (const float*)d_in[1];
    const float* mu    = (const float*)d_in[2];
    const float* lamb  = (const float*)d_in[3];
    const float* c     = (const float*)d_in[4];
    const float* beta  = (const float*)d_in[5];
    const float* u     = (const float*)d_in[6];
    float* out = (float*)d_out;

    bold_scan_kernel<<<1, 32, 0, stream>>>(mtt, sigma, mu, lamb, c, beta, u, out);
}